// MAGNNLayer_13391708029876
// MI455X (gfx1250) — compile-verified
//
#include <hip/hip_runtime.h>
#include <hip/hip_bf16.h>

#define M_   4
#define NI_  50000
#define L_   4
#define T_   3
#define N_   20000
#define IN_  128
#define OUT_ 64
#define E_   4
#define WPB  4            // waves per block in encoder kernel
#define NTILES (NI_/16)   // 3125, exact

typedef __attribute__((ext_vector_type(2))) float v2f;
typedef __attribute__((ext_vector_type(8))) float v8f;

// ---------------------------------------------------------------------------
// Kernel 1: gather + encode (WMMA f32 16x16x4) + bias + attention scores
//   grid = (ceil(NTILES/WPB), M), block = 32*WPB
//   W_enc[m] staged into LDS interleaved by K-pairs so each B fragment
//   ({W[k][c], W[k+1][c]}) is one aligned ds_load_b64 into an even VGPR pair.
// ---------------------------------------------------------------------------
__global__ __launch_bounds__(32*WPB)
void enc_score_kernel(const float* __restrict__ feats,
                      const float* __restrict__ W_enc,
                      const float* __restrict__ b_enc,
                      const float* __restrict__ W_att,
                      const float* __restrict__ b_att,
                      const int*   __restrict__ inst_types,
                      const int*   __restrict__ inst_ids,
                      const int*   __restrict__ edge_types,
                      float* __restrict__ lastbuf,   // [M, NI, OUT]
                      float* __restrict__ scores)    // [M, NI]
{
    // Wp2[p*OUT_ + c] = { W[2p][c], W[2p+1][c] }  -- 64 pairs x 64 cols = 32 KB
    __shared__ v2f Wp2[(IN_/2) * OUT_];

    const int m = blockIdx.y;

    // cooperative interleaved stage of W_enc[m] into LDS
    {
        const float* Wg = W_enc + (size_t)m * IN_ * OUT_;
        for (int i = threadIdx.x; i < (IN_/2) * OUT_; i += blockDim.x) {
            const int p = i >> 6;       // K-pair index
            const int c = i & 63;       // output column
            v2f w;
            w.x = Wg[(2*p)     * OUT_ + c];
            w.y = Wg[(2*p + 1) * OUT_ + c];
            Wp2[i] = w;
        }
    }
    __syncthreads();

    const int lane = threadIdx.x & 31;
    const int wave = threadIdx.x >> 5;
    const int tile = blockIdx.x * WPB + wave;
    if (tile >= NTILES) return;          // wave-uniform branch; EXEC stays all-1

    const int base = tile * 16;
    const int r    = lane & 15;          // A row within tile / B column in group
    const int hi   = lane >> 4;          // half-wave: 0 -> K+0,1 ; 1 -> K+2,3
    const int koff = hi * 2;
    const int n    = base + r;

    // gather row pointers for l=0 (first) and l=3 (last)
    const size_t ib = ((size_t)m * NI_ + n) * L_;
    const int t0 = inst_types[ib + 0], id0 = inst_ids[ib + 0];
    const int t3 = inst_types[ib + 3], id3 = inst_ids[ib + 3];
    const float* p0 = feats + ((size_t)t0 * N_ + id0) * IN_ + koff;
    const float* p3 = feats + ((size_t)t3 * N_ + id3) * IN_ + koff;

    // per-lane B base: pair (2k + hi), column r; advance 2*OUT_ v2f per k-step
    const v2f* wbase = Wp2 + hi * OUT_ + r;

    v8f accF[4] = {};   // first: 4 N-tiles of 16 cols
    v8f accL[4] = {};   // last

    #pragma unroll 4
    for (int k = 0; k < IN_ / 4; ++k) {
        // A fragments: 16x4 fp32 (lane&15 = M, half-wave selects K pair)
        v2f a0 = *(const v2f*)(p0 + 4 * k);
        v2f a3 = *(const v2f*)(p3 + 4 * k);

        const v2f* wk = wbase + (size_t)k * (2 * OUT_);
        #pragma unroll
        for (int t = 0; t < 4; ++t) {
            v2f b = wk[16 * t];          // single ds_load_b64 per fragment
            accF[t] = __builtin_amdgcn_wmma_f32_16x16x4_f32(
                false, a0, false, b, (short)0, accF[t], false, false);
            accL[t] = __builtin_amdgcn_wmma_f32_16x16x4_f32(
                false, a3, false, b, (short)0, accL[t], false, false);
        }
    }

    // bias + attention score partials + store `last`
    const int    et = edge_types[m];
    const float* aW = W_att + (size_t)et * (2 * OUT_);
    const float  ab = b_att[et];

    float partial[8];
    #pragma unroll
    for (int vr = 0; vr < 8; ++vr) partial[vr] = 0.0f;

    #pragma unroll
    for (int t = 0; t < 4; ++t) {
        const int   col  = 16 * t + r;
        const float bias = b_enc[m * OUT_ + col];
        const float wf   = aW[col];
        const float wl   = aW[OUT_ + col];
        #pragma unroll
        for (int vr = 0; vr < 8; ++vr) {
            // D layout: VGPR vr holds row vr (lanes 0-15) / row 8+vr (lanes 16-31)
            const float f = accF[t][vr] + bias;
            const float l = accL[t][vr] + bias;
            partial[vr] += f * wf + l * wl;
            const int row = hi * 8 + vr;
            lastbuf[((size_t)m * NI_ + base + row) * OUT_ + col] = l;
        }
    }

    // reduce score partials across the 16 lanes of each half-wave
    #pragma unroll
    for (int mask = 1; mask < 16; mask <<= 1) {
        #pragma unroll
        for (int vr = 0; vr < 8; ++vr)
            partial[vr] += __shfl_xor(partial[vr], mask, 32);
    }
    if (r == 0) {
        #pragma unroll
        for (int vr = 0; vr < 8; ++vr) {
            float s = partial[vr] + ab;
            s = (s > 0.0f) ? s : 0.2f * s;          // leaky_relu(0.2)
            scores[(size_t)m * NI_ + base + hi * 8 + vr] = s;
        }
    }
}

// ---------------------------------------------------------------------------
// Kernel 2: per-metapath softmax statistics (max, sum of exp) over NI scores
// ---------------------------------------------------------------------------
__global__ __launch_bounds__(256)
void softmax_stats_kernel(const float* __restrict__ scores,
                          float* __restrict__ stats)   // [M][2] = {max, sumexp}
{
    __shared__ float red[256];
    const int m   = blockIdx.x;
    const int tid = threadIdx.x;

    float mx = -3.4e38f;
    for (int nn = tid; nn < NI_; nn += 256)
        mx = fmaxf(mx, scores[(size_t)m * NI_ + nn]);
    red[tid] = mx; __syncthreads();
    for (int s = 128; s > 0; s >>= 1) {
        if (tid < s) red[tid] = fmaxf(red[tid], red[tid + s]);
        __syncthreads();
    }
    mx = red[0]; __syncthreads();

    float sum = 0.0f;
    for (int nn = tid; nn < NI_; nn += 256)
        sum += __expf(scores[(size_t)m * NI_ + nn] - mx);
    red[tid] = sum; __syncthreads();
    for (int s = 128; s > 0; s >>= 1) {
        if (tid < s) red[tid] += red[tid + s];
        __syncthreads();
    }
    if (tid == 0) { stats[2 * m] = mx; stats[2 * m + 1] = red[0]; }
}

// ---------------------------------------------------------------------------
// Kernel 3: mp_out[m][o] = sum_n softmax_w[n] * last[m][n][o]
//   block = 512 threads = 8 row-groups x 64 columns (coalesced rows of last)
// ---------------------------------------------------------------------------
__global__ __launch_bounds__(512)
void mp_out_kernel(const float* __restrict__ lastbuf,
                   const float* __restrict__ scores,
                   const float* __restrict__ stats,
                   float* __restrict__ mp)             // [M, OUT]
{
    __shared__ float red[512];
    const int m   = blockIdx.x;
    const int tid = threadIdx.x;
    const int o   = tid & 63;
    const int g   = tid >> 6;

    const float mx  = stats[2 * m];
    const float inv = 1.0f / stats[2 * m + 1];

    float acc = 0.0f;
    for (int nn = g; nn < NI_; nn += 8) {
        const float w = __expf(scores[(size_t)m * NI_ + nn] - mx);
        acc += w * lastbuf[((size_t)m * NI_ + nn) * OUT_ + o];
    }
    red[tid] = acc; __syncthreads();
    for (int s = 256; s >= 64; s >>= 1) {
        if (tid < s) red[tid] += red[tid + s];
        __syncthreads();
    }
    if (tid < 64) mp[m * OUT_ + tid] = red[tid] * inv;
}

// ---------------------------------------------------------------------------
// Kernel 4: metapath-level softmax + ELU -> 64 outputs (single tiny block)
// ---------------------------------------------------------------------------
__global__ __launch_bounds__(64)
void final_kernel(const float* __restrict__ mp,
                  const float* __restrict__ w_mp,
                  const float* __restrict__ b_mp,
                  float* __restrict__ out)
{
    __shared__ float red[64];
    __shared__ float msc[M_];
    const int o = threadIdx.x;
    const float wv = w_mp[o];

    for (int m = 0; m < M_; ++m) {
        red[o] = mp[m * OUT_ + o] * wv;
        __syncthreads();
        for (int s = 32; s > 0; s >>= 1) {
            if (o < s) red[o] += red[o + s];
            __syncthreads();
        }
        if (o == 0) {
            float x = red[0] + b_mp[0];
            msc[m] = (x > 0.0f) ? x : 0.2f * x;     // leaky_relu(0.2)
        }
        __syncthreads();
    }

    const float mmax = fmaxf(fmaxf(msc[0], msc[1]), fmaxf(msc[2], msc[3]));
    const float e0 = __expf(msc[0] - mmax);
    const float e1 = __expf(msc[1] - mmax);
    const float e2 = __expf(msc[2] - mmax);
    const float e3 = __expf(msc[3] - mmax);
    const float den = e0 + e1 + e2 + e3;

    float v = (e0 * mp[0 * OUT_ + o] + e1 * mp[1 * OUT_ + o] +
               e2 * mp[2 * OUT_ + o] + e3 * mp[3 * OUT_ + o]) / den;
    out[o] = (v > 0.0f) ? v : (__expf(v) - 1.0f);   // elu
}

// ---------------------------------------------------------------------------
extern "C" void kernel_launch(void* const* d_in, const int* in_sizes, int n_in,
                              void* d_out, int out_size, void* d_ws, size_t ws_size,
                              hipStream_t stream)
{
    const float* feats      = (const float*)d_in[0];
    const float* W_enc      = (const float*)d_in[1];
    const float* b_enc      = (const float*)d_in[2];
    const float* W_att      = (const float*)d_in[3];
    const float* b_att      = (const float*)d_in[4];
    const float* w_mp       = (const float*)d_in[5];
    const float* b_mp       = (const float*)d_in[6];
    const int*   inst_types = (const int*)d_in[7];
    const int*   inst_ids   = (const int*)d_in[8];
    const int*   edge_types = (const int*)d_in[9];
    float*       out        = (float*)d_out;

    // workspace layout (floats): last[M*NI*OUT] | scores[M*NI] | stats[2M] | mp[M*OUT]
    float* ws      = (float*)d_ws;
    float* lastbuf = ws;
    float* scores  = lastbuf + (size_t)M_ * NI_ * OUT_;
    float* stats   = scores  + (size_t)M_ * NI_;
    float* mp      = stats   + 2 * M_;

    dim3 g1((NTILES + WPB - 1) / WPB, M_);
    enc_score_kernel<<<g1, 32 * WPB, 0, stream>>>(
        feats, W_enc, b_enc, W_att, b_att, inst_types, inst_ids, edge_types,
        lastbuf, scores);

    softmax_stats_kernel<<<M_, 256, 0, stream>>>(scores, stats);
    mp_out_kernel<<<M_, 512, 0, stream>>>(lastbuf, scores, stats, mp);
    final_kernel<<<1, 64, 0, stream>>>(mp, w_mp, b_mp, out);
}